// SCSA_10531259810261
// MI455X (gfx1250) — compile-verified
//
#include <hip/hip_runtime.h>
#include <cfloat>

typedef float v2f __attribute__((ext_vector_type(2)));
typedef float v8f __attribute__((ext_vector_type(8)));

// D = A(16x4,f32) * B(4x16,f32) + C(16x16,f32)  -- CDNA5 V_WMMA_F32_16X16X4_F32
__device__ __forceinline__ v8f wmma4(v2f a, v2f b, v8f c) {
  return __builtin_amdgcn_wmma_f32_16x16x4_f32(false, a, false, b, (short)0, c, false, false);
}

// ---------------------------------------------------------------------------
// K1: rinv[s] = 1/||x[:,:,s]||_F  over (B,C), s = h*32+w  (1024 spatial pos)
// x flat: [(b*512+c)*1024 + s]
// ---------------------------------------------------------------------------
__global__ void norm_kernel(const float* __restrict__ x, float* __restrict__ rinv) {
  __shared__ float red[256];
  const int s = blockIdx.x;
  float acc = 0.f;
  for (int bc = threadIdx.x; bc < 8192; bc += 256) {
    const float v = x[(size_t)bc * 1024 + s];
    acc += v * v;
  }
  red[threadIdx.x] = acc;
  __syncthreads();
  for (int o = 128; o > 0; o >>= 1) {
    if (threadIdx.x < o) red[threadIdx.x] += red[threadIdx.x + o];
    __syncthreads();
  }
  if (threadIdx.x == 0) rinv[s] = rsqrtf(red[0]);
}

// ---------------------------------------------------------------------------
// K2: XP[b][m][c] = x[b][c][m] * rinv[m]   (tiled transpose, both sides coalesced)
// ---------------------------------------------------------------------------
__global__ void build_xp_kernel(const float* __restrict__ x,
                                const float* __restrict__ rinv,
                                float* __restrict__ XP) {
  __shared__ float tile[32][33];
  const int c0 = blockIdx.x * 32;   // 16 tiles over C=512
  const int m0 = blockIdx.y * 32;   // 32 tiles over HW=1024
  const int b  = blockIdx.z;        // 16
  const int tx = threadIdx.x & 31;
  const int ty = threadIdx.x >> 5;  // 0..7
  for (int i = ty; i < 32; i += 8)
    tile[i][tx] = x[((size_t)(b * 512 + c0 + i)) * 1024 + m0 + tx] * rinv[m0 + tx];
  __syncthreads();
  for (int i = ty; i < 32; i += 8)
    XP[((size_t)(b * 1024 + m0 + i)) * 512 + c0 + tx] = tile[tx][i];
}

// ---------------------------------------------------------------------------
// K3: C[M,N] = A[M,K] * Bw[N,K]^T + bias[N]  (one wave per 16x16 tile, f32 WMMA)
// ---------------------------------------------------------------------------
__global__ void gemm_bias_kernel(const float* __restrict__ A,
                                 const float* __restrict__ Bw,
                                 const float* __restrict__ bias,
                                 float* __restrict__ C,
                                 int M, int N, int K) {
  const int lane   = threadIdx.x & 31;
  const int wave   = threadIdx.x >> 5;
  const int tileId = blockIdx.x * (blockDim.x >> 5) + wave;
  const int tilesN = N >> 4;
  const int tm = (tileId / tilesN) << 4;
  const int tn = (tileId % tilesN) << 4;
  if (tm >= M) return;
  const int half = lane >> 4;  // K sub-pair select (A: K0/1 vs K2/3)
  const int l16  = lane & 15;
  const float* Arow = A  + (size_t)(tm + l16) * K;
  const float* Brow = Bw + (size_t)(tn + l16) * K;
  v8f acc = {};
  for (int k = 0; k < K; k += 4) {
    const int kb = k + (half << 1);
    v2f a, b;
    a.x = Arow[kb]; a.y = Arow[kb + 1];
    b.x = Brow[kb]; b.y = Brow[kb + 1];
    acc = wmma4(a, b, acc);
  }
  const float bv = bias[tn + l16];
#pragma unroll
  for (int r = 0; r < 8; ++r) {
    const int om = tm + r + (half << 3);
    C[(size_t)om * N + tn + l16] = acc[r] + bv;
  }
}

// ---------------------------------------------------------------------------
// K4 (fused attention): per (b, 16-row tile):
//   S[16,1024] = F_tile * G^T (B-columns permuted n -> (n%32)*32 + n/32)  [LDS]
//   softmax rows (exp left in LDS, 1/sum in rowstat)
//   FDD = scale * (beta @ H) + XP
// 256 threads (8 waves). Dynamic LDS: 16*1024 + 128 + 16 floats.
// ---------------------------------------------------------------------------
__global__ void attn_kernel(const float* __restrict__ Fm,
                            const float* __restrict__ Gm,
                            const float* __restrict__ Hm,
                            const float* __restrict__ XP,
                            const float* __restrict__ scale,
                            float* __restrict__ FDD) {
  extern __shared__ float smem[];
  float* sS      = smem;              // 16*1024 scores
  float* red     = smem + 16 * 1024;  // 8*16 cross-wave partials
  float* rowstat = red + 128;         // 16: row max, then 1/rowsum

  const int b    = blockIdx.x >> 6;
  const int mt   = (blockIdx.x & 63) << 4;
  const int lane = threadIdx.x & 31;
  const int wave = threadIdx.x >> 5;  // 0..7
  const int half = lane >> 4;
  const int l16  = lane & 15;

  // ---- phase 1: scores into LDS (each wave: 8 column tiles, K=64) ----
  const float* Fr = Fm + ((size_t)b * 1024 + mt + l16) * 64;
#pragma unroll
  for (int t = 0; t < 8; ++t) {
    const int tn = ((wave << 3) + t) << 4;
    const int n  = tn + l16;
    const int pn = ((n & 31) << 5) | (n >> 5);  // W-major key permutation
    const float* Gr = Gm + ((size_t)b * 1024 + pn) * 64;
    v8f acc = {};
#pragma unroll
    for (int k = 0; k < 64; k += 4) {
      const int kb = k + (half << 1);
      v2f a, bb;
      a.x  = Fr[kb]; a.y  = Fr[kb + 1];
      bb.x = Gr[kb]; bb.y = Gr[kb + 1];
      acc = wmma4(a, bb, acc);
    }
#pragma unroll
    for (int r = 0; r < 8; ++r)
      sS[(r + (half << 3)) * 1024 + n] = acc[r];
  }
  __syncthreads();

  // ---- phase 2: row softmax stats (thread: row = tid&15, segment = tid>>4) --
  const int rowr = threadIdx.x & 15;
  const int seg  = threadIdx.x >> 4;
  float* rp = sS + rowr * 1024 + seg * 64;
  float mx = -FLT_MAX;
  for (int j = 0; j < 64; ++j) mx = fmaxf(mx, rp[j]);
  mx = fmaxf(mx, __shfl_xor(mx, 16));
  if (lane < 16) red[wave * 16 + lane] = mx;
  __syncthreads();
  if (threadIdx.x < 16) {
    float m = red[threadIdx.x];
    for (int wv = 1; wv < 8; ++wv) m = fmaxf(m, red[wv * 16 + threadIdx.x]);
    rowstat[threadIdx.x] = m;
  }
  __syncthreads();
  const float rmax = rowstat[rowr];
  float ssum = 0.f;
  for (int j = 0; j < 64; ++j) {
    const float e = __expf(rp[j] - rmax);
    rp[j] = e;
    ssum += e;
  }
  ssum += __shfl_xor(ssum, 16);
  __syncthreads();
  if (lane < 16) red[wave * 16 + lane] = ssum;
  __syncthreads();
  if (threadIdx.x < 16) {
    float tsum = 0.f;
    for (int wv = 0; wv < 8; ++wv) tsum += red[wv * 16 + threadIdx.x];
    rowstat[threadIdx.x] = 1.0f / tsum;
  }
  __syncthreads();

  // ---- phase 3: O = beta @ H (K=1024), epilogue fdd = scale*O/rowsum + XP ---
  const float sc = scale[0];
  const float* Hb  = Hm + (size_t)b * 1024 * 512;
  const float* XPb = XP + ((size_t)b * 1024 + mt) * 512;
  float* Ob = FDD + ((size_t)b * 1024 + mt) * 512;
  const int tn0 = wave << 6;  // 64 output columns per wave (4 tiles)
  v8f acc0 = {}, acc1 = {}, acc2 = {}, acc3 = {};
  for (int k = 0; k < 1024; k += 4) {
    const int kb = k + (half << 1);
    v2f a;
    a.x = sS[l16 * 1024 + kb];
    a.y = sS[l16 * 1024 + kb + 1];
    const float* h0 = Hb + (size_t)kb * 512 + tn0 + l16;
    v2f b0, b1, b2, b3;
    b0.x = h0[0];  b0.y = h0[512];
    b1.x = h0[16]; b1.y = h0[16 + 512];
    b2.x = h0[32]; b2.y = h0[32 + 512];
    b3.x = h0[48]; b3.y = h0[48 + 512];
    acc0 = wmma4(a, b0, acc0);
    acc1 = wmma4(a, b1, acc1);
    acc2 = wmma4(a, b2, acc2);
    acc3 = wmma4(a, b3, acc3);
  }
#pragma unroll
  for (int r = 0; r < 8; ++r) {
    const int mrow = r + (half << 3);
    const float inv = rowstat[mrow];
    const size_t base = (size_t)mrow * 512 + tn0 + l16;
    Ob[base +  0] = sc * (acc0[r] * inv) + XPb[base +  0];
    Ob[base + 16] = sc * (acc1[r] * inv) + XPb[base + 16];
    Ob[base + 32] = sc * (acc2[r] * inv) + XPb[base + 32];
    Ob[base + 48] = sc * (acc3[r] * inv) + XPb[base + 48];
  }
}

// ---------------------------------------------------------------------------
// K5 (fused corr + top-48): corr[m,n] = sum_c FDD[b,m*512+c] * FDD[b,c*1024+n]
// (same buffer, two stride views). Tile [16,1024] in LDS, then one wave per
// row extracts top-48 by iterative masked wave-max. relu applied on store.
// 512 threads (16 waves). Dynamic LDS: 16*1024 floats.
// ---------------------------------------------------------------------------
__global__ void corr_topk_kernel(const float* __restrict__ FDD,
                                 float* __restrict__ TOP) {
  extern __shared__ float smem[];
  float* cS = smem;  // 16*1024
  const int b    = blockIdx.x >> 6;
  const int mt   = (blockIdx.x & 63) << 4;
  const int lane = threadIdx.x & 31;
  const int wave = threadIdx.x >> 5;  // 0..15
  const int half = lane >> 4;
  const int l16  = lane & 15;
  const float* Fb = FDD + (size_t)b * 524288;
  const float* Ar = Fb + (size_t)(mt + l16) * 512;
  const int tn0 = wave << 6;  // 64 columns per wave (4 tiles)
  v8f acc0 = {}, acc1 = {}, acc2 = {}, acc3 = {};
  for (int k = 0; k < 512; k += 4) {
    const int kb = k + (half << 1);
    v2f a;
    a.x = Ar[kb]; a.y = Ar[kb + 1];
    const float* r0 = Fb + (size_t)kb * 1024 + tn0 + l16;
    v2f b0, b1, b2, b3;
    b0.x = r0[0];  b0.y = r0[1024];
    b1.x = r0[16]; b1.y = r0[16 + 1024];
    b2.x = r0[32]; b2.y = r0[32 + 1024];
    b3.x = r0[48]; b3.y = r0[48 + 1024];
    acc0 = wmma4(a, b0, acc0);
    acc1 = wmma4(a, b1, acc1);
    acc2 = wmma4(a, b2, acc2);
    acc3 = wmma4(a, b3, acc3);
  }
#pragma unroll
  for (int r = 0; r < 8; ++r) {
    const int mr = r + (half << 3);
    cS[mr * 1024 + tn0 + l16 +  0] = acc0[r];
    cS[mr * 1024 + tn0 + l16 + 16] = acc1[r];
    cS[mr * 1024 + tn0 + l16 + 32] = acc2[r];
    cS[mr * 1024 + tn0 + l16 + 48] = acc3[r];
  }
  __syncthreads();

  // top-48 per row: wave w owns row w; 32 values per lane
  float* rowp = cS + wave * 1024;
  float* outp = TOP + ((size_t)b * 1024 + mt + wave) * 48;
  for (int t = 0; t < 48; ++t) {
    float bv = -FLT_MAX;
    int   bi = lane;
    for (int j = 0; j < 32; ++j) {
      const int idx = lane + (j << 5);
      const float v = rowp[idx];
      if (v > bv) { bv = v; bi = idx; }
    }
#pragma unroll
    for (int off = 16; off > 0; off >>= 1) {
      const float ov = __shfl_xor(bv, off);
      const int   oi = __shfl_xor(bi, off);
      if (ov > bv || (ov == bv && oi < bi)) { bv = ov; bi = oi; }
    }
    if (lane == 0) {
      outp[t]  = fmaxf(bv, 0.0f);  // relu(topk) == topk-then-relu on values
      rowp[bi] = -FLT_MAX;         // mask extracted element
    }
    __syncthreads();               // uniform (all 16 waves run 48 iterations)
  }
}

// ---------------------------------------------------------------------------
// K6: CINV[w*48+t] = rsqrt( sum_{b,h} TOP[b, h*32+w, t]^2 )
// ---------------------------------------------------------------------------
__global__ void colnorm_kernel(const float* __restrict__ TOP,
                               float* __restrict__ CINV) {
  const int idx = blockIdx.x * 256 + threadIdx.x;
  if (idx >= 32 * 48) return;
  const int w = idx / 48, t = idx % 48;
  float s = 0.f;
  for (int b = 0; b < 16; ++b)
    for (int h = 0; h < 32; ++h) {
      const float v = TOP[((size_t)b * 1024 + h * 32 + w) * 48 + t];
      s += v * v;
    }
  CINV[idx] = rsqrtf(s);
}

// ---------------------------------------------------------------------------
// K7: out[b][t][h][w] = TOP[b][h*32+w][t] * CINV[w*48+t]
// ---------------------------------------------------------------------------
__global__ void final_kernel(const float* __restrict__ TOP,
                             const float* __restrict__ CINV,
                             float* __restrict__ out) {
  const int idx = blockIdx.x * 256 + threadIdx.x;  // 786432 = 16*48*32*32
  const int w = idx & 31;
  const int h = (idx >> 5) & 31;
  const int rest = idx >> 10;       // b*48 + t
  const int t = rest % 48;
  const int b = rest / 48;
  out[idx] = TOP[((size_t)b * 1024 + h * 32 + w) * 48 + t] * CINV[w * 48 + t];
}

// ---------------------------------------------------------------------------
extern "C" void kernel_launch(void* const* d_in, const int* in_sizes, int n_in,
                              void* d_out, int out_size, void* d_ws, size_t ws_size,
                              hipStream_t stream) {
  (void)in_sizes; (void)n_in; (void)out_size; (void)ws_size;
  const float* x     = (const float*)d_in[0];
  const float* fW    = (const float*)d_in[1];
  const float* fb    = (const float*)d_in[2];
  const float* gW    = (const float*)d_in[3];
  const float* gb    = (const float*)d_in[4];
  const float* hW    = (const float*)d_in[5];
  const float* hb    = (const float*)d_in[6];
  const float* scale = (const float*)d_in[7];
  float* out = (float*)d_out;

  // workspace layout (floats), total ~112 MB
  float* w    = (float*)d_ws;
  float* rinv = w;                         // 1024
  float* XP   = rinv + 1024;               // 16*1024*512 = 8388608
  float* Fm   = XP + (size_t)8388608;      // 16*1024*64  = 1048576
  float* Gm   = Fm + (size_t)1048576;      // 1048576
  float* Hm   = Gm + (size_t)1048576;      // 8388608
  float* FDD  = Hm + (size_t)8388608;      // 8388608
  float* TOPb = FDD + (size_t)8388608;     // 16*1024*48 = 786432
  float* CINV = TOPb + (size_t)786432;     // 1536

  // allow >64KB dynamic LDS (CDNA5: up to 320KB per workgroup)
  hipFuncSetAttribute((const void*)attn_kernel,
                      hipFuncAttributeMaxDynamicSharedMemorySize, 66112);
  hipFuncSetAttribute((const void*)corr_topk_kernel,
                      hipFuncAttributeMaxDynamicSharedMemorySize, 65536);

  norm_kernel<<<1024, 256, 0, stream>>>(x, rinv);
  build_xp_kernel<<<dim3(16, 32, 16), 256, 0, stream>>>(x, rinv, XP);
  gemm_bias_kernel<<<512, 256, 0, stream>>>(XP, fW, fb, Fm, 16384, 64, 512);
  gemm_bias_kernel<<<512, 256, 0, stream>>>(XP, gW, gb, Gm, 16384, 64, 512);
  gemm_bias_kernel<<<4096, 256, 0, stream>>>(XP, hW, hb, Hm, 16384, 512, 512);
  attn_kernel<<<1024, 256, 66112, stream>>>(Fm, Gm, Hm, XP, scale, FDD);
  corr_topk_kernel<<<1024, 512, 65536, stream>>>(FDD, TOPb);
  colnorm_kernel<<<6, 256, 0, stream>>>(TOPb, CINV);
  final_kernel<<<3072, 256, 0, stream>>>(TOPb, CINV, out);
}